// KalmanFilter_5205500363211
// MI455X (gfx1250) — compile-verified
//
#include <hip/hip_runtime.h>

#define GDIM   4096
#define TSTEPS 200
#define MD     8
#define SD     32
#define WAVES_PER_BLK 4

typedef __attribute__((ext_vector_type(2))) float v2f;
typedef __attribute__((ext_vector_type(8))) float v8f;

// ---- WMMA f32 16x16x4 fragment gathers (ISA 7.12.2 layouts) ----
// A (16x4, MxK): lane l -> row = rowBase + (l&15), k0 = kBase + ((l>>4)<<1)
// k0 is even -> 8B-aligned -> single ds_load_b64 into an aligned VGPR pair.
__device__ __forceinline__ v2f fragA_rowmajor(const float* buf, int rowBase, int kBase, int lane) {
  const int r  = rowBase + (lane & 15);
  const int k0 = kBase + ((lane >> 4) << 1);
  return *(const v2f*)(buf + r * SD + k0);
}
// B (4x16, KxN) from row-major buf (B[k][n] = buf[k*32+n]) -- rows are 128B apart (strided)
__device__ __forceinline__ v2f fragB_rowmajor(const float* buf, int kBase, int colBase, int lane) {
  const int n  = colBase + (lane & 15);
  const int k0 = kBase + ((lane >> 4) << 1);
  v2f b;
  b.x = buf[k0 * SD + n];
  b.y = buf[(k0 + 1) * SD + n];
  return b;
}
// B = buf^T : B[k][n] = buf[n*32+k] -- contiguous pair -> ds_load_b64
__device__ __forceinline__ v2f fragB_transposed(const float* buf, int kBase, int colBase, int lane) {
  const int n  = colBase + (lane & 15);
  const int k0 = kBase + ((lane >> 4) << 1);
  return *(const v2f*)(buf + n * SD + k0);
}
// A from transposed source: A[m][k] = src[k][m]  (strided; used for Kneg = Xneg^T)
__device__ __forceinline__ v2f fragA_transposed(const float* buf, int rowBase, int kBase, int lane) {
  const int m  = rowBase + (lane & 15);
  const int k0 = kBase + ((lane >> 4) << 1);
  v2f a;
  a.x = buf[k0 * SD + m];
  a.y = buf[(k0 + 1) * SD + m];
  return a;
}

#define WMMA_F32(A, B, C) \
  __builtin_amdgcn_wmma_f32_16x16x4_f32(false, (A), false, (B), (short)0, (C), false, false)

__global__ __launch_bounds__(WAVES_PER_BLK * 32)
void kalman_wmma_kernel(const float* __restrict__ xg,
                        const float* __restrict__ Fg,
                        const float* __restrict__ Hg,
                        const float* __restrict__ Qg,
                        const float* __restrict__ Rg,
                        const float* __restrict__ m0g,
                        const float* __restrict__ P0g,
                        float* __restrict__ outMeans,
                        float* __restrict__ outCovs)
{
  // shared (read-only) model matrices
  __shared__ alignas(16) float sF[SD * SD];
  __shared__ alignas(16) float sQ[SD * SD];
  __shared__ alignas(16) float sHpad[16 * SD];   // H padded to 16 rows (rows 8..15 zero)
  __shared__ alignas(16) float sR[MD * MD];
  // per-wave filter state / scratch
  __shared__ alignas(16) float sP   [WAVES_PER_BLK][SD * SD];
  __shared__ alignas(16) float sPost[WAVES_PER_BLK][SD * SD];
  __shared__ alignas(16) float sHP  [WAVES_PER_BLK][16 * SD]; // HP padded (rows 8..15 = 0)
  __shared__ alignas(16) float sX   [WAVES_PER_BLK][MD * SD]; // Xneg = -(meas_cov^{-1} HP)
  __shared__ alignas(16) float sL   [WAVES_PER_BLK][MD * MD];
  __shared__ alignas(16) float smv  [WAVES_PER_BLK][SD];
  __shared__ alignas(16) float smp  [WAVES_PER_BLK][SD];
  __shared__ alignas(16) float srs  [WAVES_PER_BLK][MD];

  const int tid  = threadIdx.x;
  const int lane = tid & 31;
  const int wv   = tid >> 5;
  const int g    = blockIdx.x * WAVES_PER_BLK + wv;

  // cooperative load of shared constants
  for (int i = tid; i < SD * SD; i += WAVES_PER_BLK * 32) { sF[i] = Fg[i]; sQ[i] = Qg[i]; }
  for (int i = tid; i < 16 * SD; i += WAVES_PER_BLK * 32) sHpad[i] = (i < MD * SD) ? Hg[i] : 0.f;
  if (tid < MD * MD) sR[tid] = Rg[tid];
  // per-wave state init
  smv[wv][lane] = m0g[lane];
  for (int i = lane; i < SD * SD; i += 32) sP[wv][i] = P0g[i];
  __syncthreads();

  float* P    = sP[wv];
  float* Post = sPost[wv];
  float* HP   = sHP[wv];
  float* Xn   = sX[wv];
  float* L    = sL[wv];
  float* mv   = smv[wv];
  float* mp   = smp[wv];
  float* rs   = srs[wv];

  const int hi8 = (lane >> 4) << 3;  // 0 or 8 (C/D row offset)
  const int lo  = lane & 15;         // C/D column

  for (int t = 0; t < TSTEPS; ++t) {
    // ---- (1) predicted measurement mean + residual (lanes 0..7, VALU) ----
    if (lane < MD) {
      float mm = 0.f;
      #pragma unroll
      for (int k = 0; k < SD; ++k) mm = fmaf(sHpad[lane * SD + k], mv[k], mm);
      outMeans[((size_t)g * TSTEPS + t) * MD + lane] = mm;
      rs[lane] = xg[((size_t)g * TSTEPS + t) * MD + lane] - mm;
    }

    // ---- (2) HP = Hpad * P : WMMA, 1x2 tiles x 8 K-steps (rows 8..15 come out zero) ----
    #pragma unroll
    for (int tj = 0; tj < 2; ++tj) {
      v8f acc = {};
      #pragma unroll
      for (int kk = 0; kk < 8; ++kk) {
        v2f a = fragA_rowmajor(sHpad, 0, kk * 4, lane);
        v2f b = fragB_rowmajor(P, kk * 4, tj * 16, lane);
        acc = WMMA_F32(a, b, acc);
      }
      #pragma unroll
      for (int r = 0; r < 8; ++r)
        HP[(r + hi8) * SD + tj * 16 + lo] = acc[r];
    }

    // ---- (3) meas_cov = HP*Hpad^T + R : WMMA, 1 tile x 8 K-steps; valid 8x8 region ----
    {
      const bool valid = (hi8 == 0) && (lo < MD);
      v8f acc;
      #pragma unroll
      for (int r = 0; r < 8; ++r) {
        const float rv = sR[(r & 7) * MD + (lo & 7)];
        acc[r] = valid ? rv : 0.f;
      }
      #pragma unroll
      for (int kk = 0; kk < 8; ++kk) {
        v2f a = fragA_rowmajor(HP, 0, kk * 4, lane);
        v2f b = fragB_transposed(sHpad, kk * 4, 0, lane);   // Hpad^T
        acc = WMMA_F32(a, b, acc);
      }
      const size_t cbase = ((size_t)g * TSTEPS + t) * (size_t)(MD * MD);
      if (valid) {
        #pragma unroll
        for (int r = 0; r < 8; ++r) {
          L[r * MD + lo] = acc[r];
          outCovs[cbase + r * MD + lo] = acc[r];
        }
      }
    }

    // ---- (4) Cholesky of 8x8 SPD meas_cov (serial on lane 0) ----
    if (lane == 0) {
      for (int i = 0; i < MD; ++i) {
        float d = L[i * MD + i];
        for (int k = 0; k < i; ++k) d = fmaf(-L[i * MD + k], L[i * MD + k], d);
        d = sqrtf(fmaxf(d, 1e-30f));
        L[i * MD + i] = d;
        const float inv = 1.f / d;
        for (int r = i + 1; r < MD; ++r) {
          float v = L[r * MD + i];
          for (int k = 0; k < i; ++k) v = fmaf(-L[r * MD + k], L[i * MD + k], v);
          L[r * MD + i] = v * inv;
        }
      }
    }

    // ---- (5) Xneg = -(meas_cov^{-1} HP) : lane-per-column triangular solves ----
    {
      float y[MD];
      #pragma unroll
      for (int i = 0; i < MD; ++i) {
        float v = HP[i * SD + lane];
        #pragma unroll
        for (int k = 0; k < MD; ++k) if (k < i) v = fmaf(-L[i * MD + k], y[k], v);
        y[i] = v / L[i * MD + i];
      }
      #pragma unroll
      for (int i = MD - 1; i >= 0; --i) {
        float v = y[i];
        #pragma unroll
        for (int k = 0; k < MD; ++k) if (k > i) v = fmaf(-L[k * MD + i], y[k], v);
        y[i] = v / L[i * MD + i];
      }
      #pragma unroll
      for (int i = 0; i < MD; ++i) Xn[i * SD + lane] = -y[i];   // negated!
    }

    // ---- (6) m_post = m + K*resid  (K = -Xn^T), then m_next = F*m_post ----
    {
      float v = mv[lane];
      #pragma unroll
      for (int j = 0; j < MD; ++j) v = fmaf(-Xn[j * SD + lane], rs[j], v);
      mp[lane] = v;
      float mn = 0.f;
      #pragma unroll
      for (int k = 0; k < SD; ++k) mn = fmaf(sF[lane * SD + k], mp[k], mn);
      mv[lane] = mn;
    }

    // ---- (7) P_post = P + Kneg*HP : WMMA, 2x2 tiles x 2 K-steps, C seeded with P ----
    #pragma unroll
    for (int ti = 0; ti < 2; ++ti) {
      #pragma unroll
      for (int tj = 0; tj < 2; ++tj) {
        v8f acc;
        #pragma unroll
        for (int r = 0; r < 8; ++r)
          acc[r] = P[(ti * 16 + r + hi8) * SD + tj * 16 + lo];
        #pragma unroll
        for (int kk = 0; kk < 2; ++kk) {
          v2f a = fragA_transposed(Xn, ti * 16, kk * 4, lane);  // Kneg = Xn^T (32x8)
          v2f b = fragB_rowmajor(HP, kk * 4, tj * 16, lane);
          acc = WMMA_F32(a, b, acc);
        }
        #pragma unroll
        for (int r = 0; r < 8; ++r)
          Post[(ti * 16 + r + hi8) * SD + tj * 16 + lo] = acc[r];
      }
    }

    // ---- (8) T1 = F * P_post : WMMA 2x2 tiles x 8 K-steps; old P dead -> store T1 into P ----
    #pragma unroll
    for (int ti = 0; ti < 2; ++ti) {
      #pragma unroll
      for (int tj = 0; tj < 2; ++tj) {
        v8f acc = {};
        #pragma unroll
        for (int kk = 0; kk < 8; ++kk) {
          v2f a = fragA_rowmajor(sF, ti * 16, kk * 4, lane);
          v2f b = fragB_rowmajor(Post, kk * 4, tj * 16, lane);
          acc = WMMA_F32(a, b, acc);
        }
        #pragma unroll
        for (int r = 0; r < 8; ++r)
          P[(ti * 16 + r + hi8) * SD + tj * 16 + lo] = acc[r];
      }
    }

    // ---- (9) P_next = T1 * F^T + Q : regs until all T1 reads done, then commit to P ----
    v8f d00, d01, d10, d11;
    #pragma unroll
    for (int ti = 0; ti < 2; ++ti) {
      #pragma unroll
      for (int tj = 0; tj < 2; ++tj) {
        v8f acc;
        #pragma unroll
        for (int r = 0; r < 8; ++r)
          acc[r] = sQ[(ti * 16 + r + hi8) * SD + tj * 16 + lo];
        #pragma unroll
        for (int kk = 0; kk < 8; ++kk) {
          v2f a = fragA_rowmajor(P, ti * 16, kk * 4, lane);     // T1
          v2f b = fragB_transposed(sF, kk * 4, tj * 16, lane);  // F^T
          acc = WMMA_F32(a, b, acc);
        }
        if (ti == 0 && tj == 0)      d00 = acc;
        else if (ti == 0)            d01 = acc;
        else if (tj == 0)            d10 = acc;
        else                         d11 = acc;
      }
    }
    #pragma unroll
    for (int r = 0; r < 8; ++r) {
      P[(r + hi8) * SD + lo]            = d00[r];
      P[(r + hi8) * SD + 16 + lo]       = d01[r];
      P[(16 + r + hi8) * SD + lo]       = d10[r];
      P[(16 + r + hi8) * SD + 16 + lo]  = d11[r];
    }
  }
}

extern "C" void kernel_launch(void* const* d_in, const int* in_sizes, int n_in,
                              void* d_out, int out_size, void* d_ws, size_t ws_size,
                              hipStream_t stream) {
  const float* x  = (const float*)d_in[0];
  const float* F  = (const float*)d_in[1];
  const float* H  = (const float*)d_in[2];
  const float* Q  = (const float*)d_in[3];
  const float* R  = (const float*)d_in[4];
  const float* m0 = (const float*)d_in[5];
  const float* P0 = (const float*)d_in[6];
  float* outMeans = (float*)d_out;
  float* outCovs  = outMeans + (size_t)GDIM * TSTEPS * MD;

  dim3 grid(GDIM / WAVES_PER_BLK);
  dim3 block(WAVES_PER_BLK * 32);
  hipLaunchKernelGGL(kalman_wmma_kernel, grid, block, 0, stream,
                     x, F, H, Q, R, m0, P0, outMeans, outCovs);
}